// SAGEModel_3332894622191
// MI455X (gfx1250) — compile-verified
//
#include <hip/hip_runtime.h>
#include <cstdint>

// ---------------------------------------------------------------------------
// GraphSAGE 2-layer forward for MI455X (gfx1250).
//
// Multiply-before-aggregate: mean(x[src]) @ W_l == mean((x @ W_l)[src]),
// shrinking per-edge scatter traffic 25x (500 -> 20 floats/edge).
// The dense GEMM x[100000,500] @ [W1_l|W1_r] (padded to 48 cols) uses
// V_WMMA_F32_16X16X4_F32. x is streamed from HBM exactly once (200 MB
// ~ 8.6 us at 23.3 TB/s); the 96 KB packed weight matrix is staged in LDS
// per workgroup (pair-packed so each B fragment is one ds_load_b64), and
// each wave computes 32 rows x 48 cols so B fragments are reused across
// two A tiles (6 WMMA per 5 memory ops).
// ---------------------------------------------------------------------------

typedef __attribute__((ext_vector_type(2))) float v2f;
typedef __attribute__((ext_vector_type(8))) float v8f;

#define NNODES 100000
#define NEDGES 800000
#define F_IN   500
#define F_HID  20
#define F_OUT  3
#define WCOLS  48          // 20 (W1_l) + 20 (W1_r) + 8 zero pad -> 3 WMMA N-tiles
#define KPAIRS (F_IN / 2)  // 250 pairs of K rows (pair-packed for b64 B loads)

// ---- workspace layout (float offsets) -------------------------------------
#define OFF_DEG   0u                          // 100000  (deg, then deg_inv in place)
#define OFF_AGG1  100000u                     // 2000000 (N x 20)
#define OFF_AGG2  2100000u                    // 300000  (N x 3)
#define OFF_XW    2400000u                    // 4800000 (N x 48: [x@W1_l | x@W1_r | 0])
#define OFF_WP    7200000u                    // 24000   (250 x 48 x float2 pair-packed)
#define OFF_HL    7224000u                    // 300000  (h @ W2_l)
#define OFF_HR    7524000u                    // 300000  (h @ W2_r + b2)
// total: 7,824,000 floats = 31.3 MB of d_ws

__global__ void zero_f32_kernel(float* __restrict__ p, int n) {
  int i = blockIdx.x * blockDim.x + threadIdx.x;
  int stride = gridDim.x * blockDim.x;
  for (; i < n; i += stride) p[i] = 0.0f;
}

// Pair-packed weights: Wpp[p*48 + c] = (W[2p][c], W[2p+1][c]) where
// W = [W1_l | W1_r | 0] (500 x 48). One b64 covers both K values a lane
// needs for the 16x16x4 B fragment.
__global__ void build_wpad_kernel(const float* __restrict__ W1l,
                                  const float* __restrict__ W1r,
                                  v2f* __restrict__ Wpp) {
  int i = blockIdx.x * blockDim.x + threadIdx.x;   // over 250*48 = 12000
  if (i >= KPAIRS * WCOLS) return;
  int p = i / WCOLS, c = i - p * WCOLS;
  int k0 = 2 * p, k1 = 2 * p + 1;
  v2f v = {0.0f, 0.0f};
  if (c < F_HID) {
    v.x = W1l[k0 * F_HID + c];
    v.y = W1l[k1 * F_HID + c];
  } else if (c < 2 * F_HID) {
    v.x = W1r[k0 * F_HID + (c - F_HID)];
    v.y = W1r[k1 * F_HID + (c - F_HID)];
  }
  Wpp[i] = v;
}

__global__ void degree_kernel(const int* __restrict__ dst, float* __restrict__ deg) {
  int e = blockIdx.x * blockDim.x + threadIdx.x;
  if (e < NEDGES) atomicAdd(&deg[dst[e]], 1.0f);
}

__global__ void recip_kernel(float* __restrict__ deg) {
  int i = blockIdx.x * blockDim.x + threadIdx.x;
  if (i < NNODES) deg[i] = 1.0f / fmaxf(deg[i], 1.0f);
}

// ---- WMMA GEMM: xw[N,48] = x[N,500] @ W[500,48] ----------------------------
// 256 threads = 8 waves; each wave owns 32 rows (two 16-row A tiles) and all
// three 16-col B tiles. Weights staged in 96 KB dynamic LDS once per block.
// 100000 % 32 == 0 and 500 % 4 == 0, so live waves never need masking
// (WMMA requires EXEC all-ones; the row guard is wave-uniform).
__global__ __launch_bounds__(256)
void gemm1_wmma_kernel(const float* __restrict__ x,
                       const v2f* __restrict__ Wpp,   // [250*48] pair-packed
                       float* __restrict__ xw) {
  extern __shared__ v2f WpS[];   // 12000 * 8B = 96 KB

  // cooperative stage: global (coalesced b64) -> LDS
  for (int i = threadIdx.x; i < KPAIRS * WCOLS; i += 256)
    WpS[i] = Wpp[i];
  __syncthreads();

  const int lane = threadIdx.x & 31;
  const int wave = threadIdx.x >> 5;
  const int rowBase = (blockIdx.x * 8 + wave) * 32;
  if (rowBase >= NNODES) return;

  const int r    = lane & 15;   // A: row-in-tile / B,C: col-in-tile
  const int half = lane >> 4;   // selects K sub-pair within the 4-wide K step
  const float* xrow0 = x + (size_t)(rowBase + r) * F_IN;
  const float* xrow1 = xrow0 + (size_t)16 * F_IN;

  v8f c00 = {}, c01 = {}, c02 = {};   // rows rowBase .. +15
  v8f c10 = {}, c11 = {}, c12 = {};   // rows rowBase+16 .. +31
  #pragma unroll 5
  for (int k0 = 0; k0 < F_IN; k0 += 4) {
    const int kk = k0 + half * 2;
    // A fragments: lane holds x[row][kk..kk+1] (8B aligned: row stride 2000B,
    // kk even)
    v2f a0 = *(const v2f*)(xrow0 + kk);
    v2f a1 = *(const v2f*)(xrow1 + kk);
    // B fragments: one ds_load_b64 per tile (pair index = kk/2)
    const v2f* wp = WpS + (size_t)(kk >> 1) * WCOLS;
    v2f b0 = wp[r];
    v2f b1 = wp[r + 16];
    v2f b2 = wp[r + 32];
    c00 = __builtin_amdgcn_wmma_f32_16x16x4_f32(false, a0, false, b0, (short)0, c00, false, false);
    c01 = __builtin_amdgcn_wmma_f32_16x16x4_f32(false, a0, false, b1, (short)0, c01, false, false);
    c02 = __builtin_amdgcn_wmma_f32_16x16x4_f32(false, a0, false, b2, (short)0, c02, false, false);
    c10 = __builtin_amdgcn_wmma_f32_16x16x4_f32(false, a1, false, b0, (short)0, c10, false, false);
    c11 = __builtin_amdgcn_wmma_f32_16x16x4_f32(false, a1, false, b1, (short)0, c11, false, false);
    c12 = __builtin_amdgcn_wmma_f32_16x16x4_f32(false, a1, false, b2, (short)0, c12, false, false);
  }

  // C/D layout: lane -> col = lane&15 (+tile*16); VGPR v -> row = v + 8*(lane>>4)
  const int col  = lane & 15;
  const int rOff = half * 8;
  #pragma unroll
  for (int v = 0; v < 8; ++v) {
    float* o0 = xw + (size_t)(rowBase + rOff + v) * WCOLS;
    float* o1 = o0 + (size_t)16 * WCOLS;
    o0[col]      = c00[v];
    o0[col + 16] = c01[v];
    o0[col + 32] = c02[v];
    o1[col]      = c10[v];
    o1[col + 16] = c11[v];
    o1[col + 32] = c12[v];
  }
}

// ---- edge scatter for layer 1 (20 floats per edge) -------------------------
__global__ void scatter1_kernel(const int* __restrict__ src, const int* __restrict__ dst,
                                const float* __restrict__ xw, float* __restrict__ agg1) {
  int t = blockIdx.x * blockDim.x + threadIdx.x;
  if (t >= NEDGES * F_HID) return;
  int e = t / F_HID, f = t - e * F_HID;
  atomicAdd(&agg1[(size_t)dst[e] * F_HID + f], xw[(size_t)src[e] * WCOLS + f]);
}

// ---- fused: h = relu(deg_inv*agg1 + x@W1_r + b1); hl = h@W2_l; hr = h@W2_r + b2
__global__ __launch_bounds__(128)
void layer2_kernel(const float* __restrict__ agg1, const float* __restrict__ xw,
                   const float* __restrict__ dinv, const float* __restrict__ b1,
                   const float* __restrict__ W2l, const float* __restrict__ W2r,
                   const float* __restrict__ b2,
                   float* __restrict__ hl, float* __restrict__ hr) {
  int i = blockIdx.x * blockDim.x + threadIdx.x;
  if (i >= NNODES) return;
  const float di = dinv[i];
  const float* a  = agg1 + (size_t)i * F_HID;
  const float* xr = xw + (size_t)i * WCOLS + F_HID;   // x @ W1_r part
  float al0 = 0.f, al1 = 0.f, al2 = 0.f;
  float ar0 = 0.f, ar1 = 0.f, ar2 = 0.f;
  #pragma unroll
  for (int f = 0; f < F_HID; ++f) {
    float h = fmaxf(fmaf(a[f], di, xr[f]) + b1[f], 0.0f);
    al0 = fmaf(h, W2l[f * 3 + 0], al0);
    al1 = fmaf(h, W2l[f * 3 + 1], al1);
    al2 = fmaf(h, W2l[f * 3 + 2], al2);
    ar0 = fmaf(h, W2r[f * 3 + 0], ar0);
    ar1 = fmaf(h, W2r[f * 3 + 1], ar1);
    ar2 = fmaf(h, W2r[f * 3 + 2], ar2);
  }
  hl[(size_t)i * 3 + 0] = al0;
  hl[(size_t)i * 3 + 1] = al1;
  hl[(size_t)i * 3 + 2] = al2;
  hr[(size_t)i * 3 + 0] = ar0 + b2[0];
  hr[(size_t)i * 3 + 1] = ar1 + b2[1];
  hr[(size_t)i * 3 + 2] = ar2 + b2[2];
}

// ---- edge scatter for layer 2 (3 floats per edge) --------------------------
__global__ void scatter2_kernel(const int* __restrict__ src, const int* __restrict__ dst,
                                const float* __restrict__ hl, float* __restrict__ agg2) {
  int t = blockIdx.x * blockDim.x + threadIdx.x;
  if (t >= NEDGES * F_OUT) return;
  int e = t / F_OUT, c = t - e * F_OUT;
  atomicAdd(&agg2[(size_t)dst[e] * F_OUT + c], hl[(size_t)src[e] * F_OUT + c]);
}

__global__ void final_kernel(const float* __restrict__ agg2, const float* __restrict__ hr,
                             const float* __restrict__ dinv, float* __restrict__ out) {
  int t = blockIdx.x * blockDim.x + threadIdx.x;
  if (t >= NNODES * F_OUT) return;
  int i = t / F_OUT;
  out[t] = fmaf(agg2[t], dinv[i], hr[t]);
}

// ---------------------------------------------------------------------------
extern "C" void kernel_launch(void* const* d_in, const int* in_sizes, int n_in,
                              void* d_out, int out_size, void* d_ws, size_t ws_size,
                              hipStream_t stream) {
  const float* x    = (const float*)d_in[0];           // [100000, 500]
  const int*   ei   = (const int*)d_in[1];             // [2, 800000] row-major
  const float* W1l  = (const float*)d_in[2];           // [500, 20]
  const float* W1r  = (const float*)d_in[3];           // [500, 20]
  const float* b1   = (const float*)d_in[4];           // [20]
  const float* W2l  = (const float*)d_in[5];           // [20, 3]
  const float* W2r  = (const float*)d_in[6];           // [20, 3]
  const float* b2   = (const float*)d_in[7];           // [3]
  float*       out  = (float*)d_out;                   // [100000, 3]

  const int* src = ei;
  const int* dst = ei + NEDGES;

  float* ws   = (float*)d_ws;
  float* deg  = ws + OFF_DEG;
  float* agg1 = ws + OFF_AGG1;
  float* agg2 = ws + OFF_AGG2;
  float* xw   = ws + OFF_XW;
  v2f*   Wpp  = (v2f*)(ws + OFF_WP);
  float* hl   = ws + OFF_HL;
  float* hr   = ws + OFF_HR;

  // 1) zero accumulators (deg | agg1 | agg2 are contiguous: 2.4M floats)
  zero_f32_kernel<<<2048, 256, 0, stream>>>(ws, 2400000);
  // 2) pair-pack [W1_l | W1_r | 0] into 250 x 48 x float2
  build_wpad_kernel<<<(KPAIRS * WCOLS + 255) / 256, 256, 0, stream>>>(W1l, W1r, Wpp);
  // 3) in-degree via float atomics, then invert in place
  degree_kernel<<<(NEDGES + 255) / 256, 256, 0, stream>>>(dst, deg);
  recip_kernel<<<(NNODES + 255) / 256, 256, 0, stream>>>(deg);
  // 4) WMMA GEMM: xw = x @ W (x streamed from HBM exactly once; weights in LDS)
  gemm1_wmma_kernel<<<(NNODES + 255) / 256, 256, KPAIRS * WCOLS * sizeof(v2f), stream>>>(x, Wpp, xw);
  // 5) edge-scatter the 20-dim projected features (multiply-before-aggregate)
  scatter1_kernel<<<(NEDGES * F_HID + 255) / 256, 256, 0, stream>>>(src, dst, xw, agg1);
  // 6) fused relu + layer-2 projections (K=20, N=3 -> scalar VALU)
  layer2_kernel<<<(NNODES + 127) / 128, 128, 0, stream>>>(agg1, xw, deg, b1, W2l, W2r, b2, hl, hr);
  // 7) edge-scatter the 3-dim layer-2 messages
  scatter2_kernel<<<(NEDGES * F_OUT + 255) / 256, 256, 0, stream>>>(src, dst, hl, agg2);
  // 8) out = deg_inv * agg2 + (h @ W2_r + b2)
  final_kernel<<<(NNODES * F_OUT + 255) / 256, 256, 0, stream>>>(agg2, hr, deg, out);
}